// BoundaryLoss_22299470201558
// MI455X (gfx1250) — compile-verified
//
#include <hip/hip_runtime.h>
#include <math.h>

#define BB 8
#define HH 256
#define WW 256
#define BIGD 1000000.0f

typedef __attribute__((ext_vector_type(2))) float v2f;
typedef __attribute__((ext_vector_type(8))) float v8f;

// ---------------------------------------------------------------------------
// Kernel 1: exact 1D row distance (squared) to nearest target==1 pixel.
// One thread per row; two sequential scans (matches jax cummax formulation).
// Writes dr2[B*H*W] (float) into workspace.
// ---------------------------------------------------------------------------
__global__ void edt_row_kernel(const int* __restrict__ target,
                               float* __restrict__ dr2) {
    int r = blockIdx.x * blockDim.x + threadIdx.x;   // 0 .. B*H-1
    if (r >= BB * HH) return;
    const int* trow = target + r * WW;
    float* orow = dr2 + r * WW;

    int last = -1;
    for (int w = 0; w < WW; ++w) {
        if (trow[w] > 0) last = w;
        orow[w] = (last >= 0) ? (float)(w - last) : BIGD;
    }
    int lastr = -1;
    for (int w = WW - 1; w >= 0; --w) {
        if (trow[w] > 0) lastr = w;
        float dr_ = (lastr >= 0) ? (float)(lastr - w) : BIGD;
        float d = fminf(orow[w], dr_);
        orow[w] = d * d;   // squared row distance
    }
}

// ---------------------------------------------------------------------------
// Kernel 2: column min-plus pass (tropical matvec, not WMMA-able) + sqrt +
// sigmoid weighting + deterministic per-block LDS tree reduction.
// Grid = B*W blocks (one column each), block = H = 256 threads (8 waves).
// ---------------------------------------------------------------------------
__global__ void edt_col_kernel(const float* __restrict__ logits,
                               const float* __restrict__ dr2,
                               float* __restrict__ partials) {
    __shared__ float col[HH];
    __shared__ float red[HH];

    int b = blockIdx.x >> 8;          // W == 256
    int w = blockIdx.x & (WW - 1);
    int i = threadIdx.x;

    col[i] = dr2[(b * HH + i) * WW + w];
    __syncthreads();

    float m = 3.0e38f;
    float fi = (float)i;
#pragma unroll 8
    for (int j = 0; j < HH; ++j) {
        float diff = fi - (float)j;
        m = fminf(m, fmaf(diff, diff, col[j]));   // dr2[j] + (i-j)^2
    }
    float dist = sqrtf(m);

    float x = logits[(b * HH + i) * WW + w];
    float prob = 1.0f / (1.0f + expf(-x));
    red[i] = prob * dist;
    __syncthreads();

    // fixed-order tree reduction -> deterministic partial sum
    for (int s = HH / 2; s > 0; s >>= 1) {
        if (i < s) red[i] += red[i + s];
        __syncthreads();
    }
    if (i == 0) partials[blockIdx.x] = red[0];
}

// ---------------------------------------------------------------------------
// Kernel 3: exact f32 reduction of the 2048 block partials on the matrix pipe.
// V_WMMA_F32_16X16X4_F32 with B = all-ones: D = A x 1 + C gives exact f32
// row sums; chain C across 32 chunks of 64 partials. Single wave32, EXEC all
// ones before the WMMA loop (ISA requirement).
// ---------------------------------------------------------------------------
__global__ void final_reduce_wmma(const float* __restrict__ partials,
                                  float* __restrict__ out) {
    int lane = threadIdx.x;           // 0..31, one wave32

    v8f acc = {};                     // C/D accumulator (16x16 f32, 8 VGPRs)
    v2f ones;
    ones.x = 1.0f;
    ones.y = 1.0f;                    // B (4x16) all ones in both VGPRs

    // A layout 16x4 f32: lane L -> row M=L&15; VGPR0/1 hold K={0,1} for
    // lanes 0-15 and K={2,3} for lanes 16-31 -> contiguous float2 per lane.
    int m = lane & 15;
    int koff = (lane >> 4) << 1;

    const int nchunks = (BB * WW) / 64;   // 2048 / 64 = 32
    for (int c = 0; c < nchunks; ++c) {
        const float* p = partials + c * 64 + m * 4 + koff;
        v2f a;
        a.x = p[0];
        a.y = p[1];
        // 8 args: (neg_a, A, neg_b, B, c_mod, C, reuse_a, reuse_b)
        acc = __builtin_amdgcn_wmma_f32_16x16x4_f32(
            false, a, false, ones, (short)0, acc, false, false);
    }

    // D column 0: lane 0 holds rows 0-7 (v0..v7), lane 16 holds rows 8-15.
    float s = acc[0] + acc[1] + acc[2] + acc[3] +
              acc[4] + acc[5] + acc[6] + acc[7];
    float s_hi = __shfl(s, 16, 32);
    if (lane == 0) {
        out[0] = (s + s_hi) * (1.0f / (float)(BB * 1 * HH * WW));
    }
}

// ---------------------------------------------------------------------------
extern "C" void kernel_launch(void* const* d_in, const int* in_sizes, int n_in,
                              void* d_out, int out_size, void* d_ws, size_t ws_size,
                              hipStream_t stream) {
    const float* logits = (const float*)d_in[0];
    const int*   target = (const int*)d_in[1];
    float*       out    = (float*)d_out;

    float* dr2      = (float*)d_ws;           // B*H*W floats = 2 MB
    float* partials = dr2 + BB * HH * WW;     // B*W floats   = 8 KB

    edt_row_kernel<<<(BB * HH) / 256, 256, 0, stream>>>(target, dr2);
    edt_col_kernel<<<BB * WW, HH, 0, stream>>>(logits, dr2, partials);
    final_reduce_wmma<<<1, 32, 0, stream>>>(partials, out);
}